// HeteroGNN_49976239456887
// MI455X (gfx1250) — compile-verified
//
#include <hip/hip_runtime.h>

typedef __attribute__((ext_vector_type(2))) float v2f;
typedef __attribute__((ext_vector_type(8))) float v8f;

#define LDA 132  // 128 + 4 pad: A-fetch bank index (4*m + k) mod 64 -> 32 distinct banks

__global__ void zero_f32(float* __restrict__ p, long long n) {
  long long i = (long long)blockIdx.x * blockDim.x + threadIdx.x;
  const long long stride = (long long)gridDim.x * blockDim.x;
  for (; i < n; i += stride) p[i] = 0.0f;
}

// Fuse the two "self" linears of each SAGE layer once per layer:
//   Wsum = Wr + Lw  (128x128),  bsum = bl + lb  (128)
__global__ void prep_wsum(const float* __restrict__ Wr, const float* __restrict__ Lw,
                          const float* __restrict__ bl, const float* __restrict__ lb,
                          float* __restrict__ Wsum, float* __restrict__ bsum) {
  const int i = blockIdx.x * blockDim.x + threadIdx.x;
  if (i < 128 * 128) Wsum[i] = Wr[i] + Lw[i];
  if (i < 128) bsum[i] = bl[i] + lb[i];
}

__global__ void scatter_count(const int* __restrict__ dst, float* __restrict__ cnt, int E) {
  const int e = blockIdx.x * blockDim.x + threadIdx.x;
  if (e < E) atomicAdd(&cnt[dst[e]], 1.0f);
}

// One wave32 per edge: 128 floats = 4 per lane, gathered as float4 then atomically
// scattered into the L2-resident accumulator.
__global__ void scatter_feat(const float* __restrict__ xsrc, const int* __restrict__ src,
                             const int* __restrict__ dst, float* __restrict__ agg, int E) {
  const int lane = threadIdx.x & 31;
  const int wpb = blockDim.x >> 5;
  const int nw = gridDim.x * wpb;
  for (int e = blockIdx.x * wpb + (threadIdx.x >> 5); e < E; e += nw) {
    const int s = src[e], d = dst[e];
    const float4 v = *(const float4*)(xsrc + (long long)s * 128 + lane * 4);
    float* a = agg + (long long)d * 128 + lane * 4;
    atomicAdd(a + 0, v.x);
    atomicAdd(a + 1, v.y);
    atomicAdd(a + 2, v.z);
    atomicAdd(a + 3, v.w);
  }
}

// out[16 nodes x 128] = act( (agg/cnt) @ Wl^T + xdst @ Wsum^T + bsum )
// Block = 256 thr = 8 waves; wave w computes output columns [16w, 16w+16) via
// chained V_WMMA_F32_16X16X4_F32 over K=128.
template <bool RELU>
__global__ __launch_bounds__(256) void sage_gemm(
    const float* __restrict__ agg, const float* __restrict__ cnt,
    const float* __restrict__ xdst,
    const float* __restrict__ Wl, const float* __restrict__ Wsum,
    const float* __restrict__ bsum,
    float* __restrict__ out, int n) {
  __shared__ float sA[16 * LDA];  // mean tile
  __shared__ float sX[16 * LDA];  // x_dst tile
  const int tid = threadIdx.x;
  const int node_base = blockIdx.x * 16;

  for (int i = tid; i < 16 * 128; i += 256) {
    const int row = i >> 7, col = i & 127;
    int node = node_base + row;
    if (node >= n) node = n - 1;  // tiles divide evenly for this problem; clamp for safety
    const float ic = 1.0f / fmaxf(cnt[node], 1.0f);
    sA[row * LDA + col] = agg[(long long)node * 128 + col] * ic;
    sX[row * LDA + col] = xdst[(long long)node * 128 + col];
  }
  __syncthreads();

  const int wave = tid >> 5;
  const int lane = tid & 31;
  const int m = lane & 15;              // A row / C column index within tile
  const int khalf = (lane >> 4) << 1;   // lanes 0-15: K+{0,1}; lanes 16-31: K+{2,3}
  const int ncol = wave * 16 + m;       // B column = output feature
  const float* wl = Wl + (long long)ncol * 128;
  const float* ws = Wsum + (long long)ncol * 128;

  v8f c = {};
#pragma unroll 4
  for (int kb = 0; kb < 128; kb += 4) {
    const int k0 = kb + khalf;
    const v2f a1 = *(const v2f*)(&sA[m * LDA + k0]);
    const v2f b1 = *(const v2f*)(wl + k0);
    c = __builtin_amdgcn_wmma_f32_16x16x4_f32(false, a1, false, b1, (short)0, c, false, false);
    const v2f a2 = *(const v2f*)(&sX[m * LDA + k0]);
    const v2f b2 = *(const v2f*)(ws + k0);
    c = __builtin_amdgcn_wmma_f32_16x16x4_f32(false, a2, false, b2, (short)0, c, false, false);
  }

  // C/D layout: VGPR j, lanes 0-15 -> M=j, N=lane; lanes 16-31 -> M=j+8, N=lane-16
  const float bias = bsum[ncol];
  const int mbase = (lane >> 4) << 3;
  float* orow = out + (long long)node_base * 128 + ncol;
  if (node_base + 16 <= n) {
    // full tile: wave-uniform branch, straight-line stores (no EXEC games)
#pragma unroll
    for (int j = 0; j < 8; ++j) {
      float v = c[j] + bias;
      if (RELU) v = fmaxf(v, 0.0f);
      orow[(long long)(j + mbase) * 128] = v;
    }
  } else {
#pragma unroll
    for (int j = 0; j < 8; ++j) {
      float v = c[j] + bias;
      if (RELU) v = fmaxf(v, 0.0f);
      if (node_base + j + mbase < n) orow[(long long)(j + mbase) * 128] = v;
    }
  }
}

extern "C" void kernel_launch(void* const* d_in, const int* in_sizes, int n_in,
                              void* d_out, int out_size, void* d_ws, size_t ws_size,
                              hipStream_t stream) {
  const float* x_user = (const float*)d_in[0];
  const float* x_item = (const float*)d_in[1];
  const int* ui_src = (const int*)d_in[2];
  const int* ui_dst = (const int*)d_in[3];
  const int* iu_src = (const int*)d_in[4];
  const int* iu_dst = (const int*)d_in[5];
  const float* W1l_ui = (const float*)d_in[6];
  const float* b1_ui  = (const float*)d_in[7];
  const float* W1r_ui = (const float*)d_in[8];
  const float* W1l_iu = (const float*)d_in[9];
  const float* b1_iu  = (const float*)d_in[10];
  const float* W1r_iu = (const float*)d_in[11];
  const float* L1W_u  = (const float*)d_in[12];
  const float* L1b_u  = (const float*)d_in[13];
  const float* L1W_i  = (const float*)d_in[14];
  const float* L1b_i  = (const float*)d_in[15];
  const float* W2l_ui = (const float*)d_in[16];
  const float* b2_ui  = (const float*)d_in[17];
  const float* W2r_ui = (const float*)d_in[18];
  const float* W2l_iu = (const float*)d_in[19];
  const float* b2_iu  = (const float*)d_in[20];
  const float* W2r_iu = (const float*)d_in[21];
  const float* L2W_u  = (const float*)d_in[22];
  const float* L2b_u  = (const float*)d_in[23];
  const float* L2W_i  = (const float*)d_in[24];
  const float* L2b_i  = (const float*)d_in[25];

  const int NU = in_sizes[0] / 128;
  const int NI = in_sizes[1] / 128;
  const int E  = in_sizes[2];

  // Workspace: [agg_u | agg_i | cnt_u | cnt_i | 4x Wsum | 4x bsum]
  float* agg_u = (float*)d_ws;
  float* agg_i = agg_u + (long long)NU * 128;
  float* cnt_u = agg_i + (long long)NI * 128;
  float* cnt_i = cnt_u + NU;
  float* wsum1u = cnt_i + NI;
  float* wsum1i = wsum1u + 128 * 128;
  float* wsum2u = wsum1i + 128 * 128;
  float* wsum2i = wsum2u + 128 * 128;
  float* bsum1u = wsum2i + 128 * 128;
  float* bsum1i = bsum1u + 128;
  float* bsum2u = bsum1i + 128;
  float* bsum2i = bsum2u + 128;
  const long long agg_total = (long long)(NU + NI) * 128;
  const long long zero_total = agg_total + NU + NI;

  // h1 lives in d_out; layer-2 GEMM overwrites it in place (each block only
  // reads/writes its own 16 rows, reads complete before writes).
  float* out_u = (float*)d_out;
  float* out_i = out_u + (long long)NU * 128;
  float* h1_u = out_u;
  float* h1_i = out_i;

  const int ZB = 2048;
  const int cntGrid = (E + 255) / 256;
  const int featGrid = (E + 7) / 8;  // 8 waves/block, one edge per wave
  const int WPB = (128 * 128) / 256;

  zero_f32<<<ZB, 256, 0, stream>>>((float*)d_ws, zero_total);
  prep_wsum<<<WPB, 256, 0, stream>>>(W1r_iu, L1W_u, b1_iu, L1b_u, wsum1u, bsum1u);
  prep_wsum<<<WPB, 256, 0, stream>>>(W1r_ui, L1W_i, b1_ui, L1b_i, wsum1i, bsum1i);
  prep_wsum<<<WPB, 256, 0, stream>>>(W2r_iu, L2W_u, b2_iu, L2b_u, wsum2u, bsum2u);
  prep_wsum<<<WPB, 256, 0, stream>>>(W2r_ui, L2W_i, b2_ui, L2b_i, wsum2i, bsum2i);

  scatter_count<<<cntGrid, 256, 0, stream>>>(iu_dst, cnt_u, E);
  scatter_count<<<cntGrid, 256, 0, stream>>>(ui_dst, cnt_i, E);
  scatter_feat<<<featGrid, 256, 0, stream>>>(x_item, iu_src, iu_dst, agg_u, E);
  scatter_feat<<<featGrid, 256, 0, stream>>>(x_user, ui_src, ui_dst, agg_i, E);

  sage_gemm<true><<<(NU + 15) / 16, 256, 0, stream>>>(
      agg_u, cnt_u, x_user, W1l_iu, wsum1u, bsum1u, h1_u, NU);
  sage_gemm<true><<<(NI + 15) / 16, 256, 0, stream>>>(
      agg_i, cnt_i, x_item, W1l_ui, wsum1i, bsum1i, h1_i, NI);

  zero_f32<<<ZB, 256, 0, stream>>>((float*)d_ws, agg_total);  // counts/Wsum are reused
  scatter_feat<<<featGrid, 256, 0, stream>>>(h1_i, iu_src, iu_dst, agg_u, E);
  scatter_feat<<<featGrid, 256, 0, stream>>>(h1_u, ui_src, ui_dst, agg_i, E);

  sage_gemm<false><<<(NU + 15) / 16, 256, 0, stream>>>(
      agg_u, cnt_u, h1_u, W2l_iu, wsum2u, bsum2u, out_u, NU);
  sage_gemm<false><<<(NI + 15) / 16, 256, 0, stream>>>(
      agg_i, cnt_i, h1_i, W2l_ui, wsum2i, bsum2i, out_i, NI);
}